// EdgeScoringGNN_37572373905743
// MI455X (gfx1250) — compile-verified
//
#include <hip/hip_runtime.h>
#include <hip/hip_bf16.h>

#define CH_IN   128
#define CH_HID  64
#define BN_EPS  1e-5f

#define APITCH  132   // 128 + 4 pad: lane stride = 4 LDS banks, 8B aligned
#define BPITCH  132

typedef float v2f __attribute__((ext_vector_type(2)));
typedef float v8f __attribute__((ext_vector_type(8)));

// ---------------------------------------------------------------------------
// 1) init: deg = 1.0 (self loop), zero BN partial sums
// ---------------------------------------------------------------------------
__global__ void k_init(float* __restrict__ deg, float* __restrict__ sums,
                       float* __restrict__ sumsq, int n) {
  int i = blockIdx.x * blockDim.x + threadIdx.x;
  if (i < n) deg[i] = 1.0f;
  if (i < CH_HID) { sums[i] = 0.0f; sumsq[i] = 0.0f; }
}

// ---------------------------------------------------------------------------
// 2) degree accumulation over edges (dst side)
// ---------------------------------------------------------------------------
__global__ void k_degree(const int* __restrict__ dst, float* __restrict__ deg,
                         int nE) {
  int e = blockIdx.x * blockDim.x + threadIdx.x;
  if (e < nE) atomicAdd(&deg[dst[e]], 1.0f);
}

// ---------------------------------------------------------------------------
// 3) dis = deg^{-1/2}  (deg >= 1 always, thanks to self loop)
// ---------------------------------------------------------------------------
__global__ void k_dis(const float* __restrict__ deg, float* __restrict__ dis,
                      int n) {
  int i = blockIdx.x * blockDim.x + threadIdx.x;
  if (i < n) dis[i] = rsqrtf(deg[i]);
}

// ---------------------------------------------------------------------------
// 4) GEMM1 + fused self-loop/bias epilogue.
//    Block = 256 threads = 8 waves; block tile M=32, N=64 (full), K=128.
//    Stage A-rows (32x128) and W1^T (64x128) in LDS, then each wave does one
//    16x16 tile with 32x V_WMMA_F32_16X16X4_F32 fed by ds_load_b64.
//    A frag (16x4 f32): lanes 0-15 / 16-31 both hold M=0..15; half=lane>>4
//    selects K pair {2h, 2h+1}. B frag (4x16): N=lane&15, same K split.
//    C/D: VGPR r holds M=r (lanes 0-15) / M=8+r (lanes 16-31), N=lane&15.
//    Epilogue: hlin = C;  acc1 = C*dis[row]^2 + b1[col].
// ---------------------------------------------------------------------------
__global__ __launch_bounds__(256)
void k_gemm1_wmma(const float* __restrict__ x, const float* __restrict__ W1,
                  const float* __restrict__ dis, const float* __restrict__ b1,
                  float* __restrict__ hlin, float* __restrict__ acc1,
                  int nNodes) {
  __shared__ float sA[32 * APITCH];    // 16,896 B
  __shared__ float sBt[CH_HID * BPITCH]; // 33,792 B  (W1 transposed: [n][k])

  const int t     = threadIdx.x;
  const int mbase = blockIdx.x * 32;

  // ---- stage A: 32 rows x 128 cols, 16 floats per thread (4x float4) ----
  {
    int r = t >> 3;                 // 0..31
    int c = (t & 7) << 4;           // 0,16,...,112
    float* sp = sA + r * APITCH + c;
    if (mbase + r < nNodes) {
      const float4* gp = (const float4*)(x + (size_t)(mbase + r) * CH_IN + c);
#pragma unroll
      for (int j = 0; j < 4; ++j) ((float4*)sp)[j] = gp[j];
    } else {
#pragma unroll
      for (int j = 0; j < 16; ++j) sp[j] = 0.0f;
    }
  }
  // ---- stage W1 transposed: sBt[n][k] = W1[k*64 + n] ----
  for (int idx = t; idx < CH_IN * CH_HID; idx += 256) {
    int k = idx >> 6, n = idx & 63;
    sBt[n * BPITCH + k] = W1[idx];
  }
  __syncthreads();

  const int lane = t & 31;
  const int wid  = t >> 5;            // 0..7
  const int half = lane >> 4;         // 0/1 -> K pair select
  const int mn   = lane & 15;
  const int mt   = wid >> 2;          // 0/1 -> which 16-row half
  const int n0   = (wid & 3) << 4;    // 0,16,32,48

  const float* aP = sA  + (mt * 16 + mn) * APITCH + 2 * half;
  const float* bP = sBt + (n0 + mn) * BPITCH + 2 * half;

  v8f c = {0.f, 0.f, 0.f, 0.f, 0.f, 0.f, 0.f, 0.f};
#pragma unroll
  for (int k0 = 0; k0 < CH_IN; k0 += 4) {
    v2f a = *(const v2f*)(aP + k0);   // ds_load_b64
    v2f b = *(const v2f*)(bP + k0);   // ds_load_b64
    c = __builtin_amdgcn_wmma_f32_16x16x4_f32(
        /*neg_a=*/false, a, /*neg_b=*/false, b,
        /*c_mod=*/(short)0, c, /*reuse_a=*/false, /*reuse_b=*/false);
  }

  // ---- fused epilogue: write hlin and acc1 = hlin*dis^2 + b1 ----
  const int rowBase = mbase + mt * 16 + 8 * half;
  const int col     = n0 + mn;
  const float bias  = b1[col];
#pragma unroll
  for (int r = 0; r < 8; ++r) {
    int row = rowBase + r;
    if (row < nNodes) {
      float d = dis[row];
      size_t o = (size_t)row * CH_HID + col;
      hlin[o] = c[r];
      acc1[o] = c[r] * d * d + bias;
    }
  }
}

// ---------------------------------------------------------------------------
// 5) layer-1 edge aggregation: 16 threads/edge, 4 channels each (float4 gather)
//    acc1[dst,c] += hlin[src,c] * dis[src] * dis[dst]   (L2-resident atomics)
// ---------------------------------------------------------------------------
__global__ void k_agg1(const float* __restrict__ hlin,
                       const int* __restrict__ src, const int* __restrict__ dst,
                       const float* __restrict__ dis,
                       float* __restrict__ acc, int total) {
  int t = blockIdx.x * blockDim.x + threadIdx.x;
  if (t >= total) return;
  int e  = t >> 4;
  int c0 = (t & 15) << 2;
  int s = src[e], d = dst[e];
  float nrm = dis[s] * dis[d];
  const float4 v = *reinterpret_cast<const float4*>(hlin + (size_t)s * CH_HID + c0);
  float* o = acc + (size_t)d * CH_HID + c0;
  atomicAdd(o + 0, v.x * nrm);
  atomicAdd(o + 1, v.y * nrm);
  atomicAdd(o + 2, v.z * nrm);
  atomicAdd(o + 3, v.w * nrm);
}

// ---------------------------------------------------------------------------
// 6) BN statistics: per-channel partial sums, then global atomics
// ---------------------------------------------------------------------------
__global__ void k_bnstats(const float* __restrict__ acc,
                          float* __restrict__ sums, float* __restrict__ sumsq,
                          int n) {
  int tid  = blockIdx.x * blockDim.x + threadIdx.x;
  int c    = tid & 63;
  int grp  = tid >> 6;
  int ngrp = (gridDim.x * blockDim.x) >> 6;
  float s = 0.f, q = 0.f;
  for (int i = grp; i < n; i += ngrp) {
    float v = acc[(size_t)i * CH_HID + c];
    s += v; q += v * v;
  }
  atomicAdd(&sums[c], s);
  atomicAdd(&sumsq[c], q);
}

// ---------------------------------------------------------------------------
// 7) BN finalize: fold mean/var/gamma/beta into scale/shift  (1 block, 64 thr)
// ---------------------------------------------------------------------------
__global__ void k_bnfinal(const float* __restrict__ sums,
                          const float* __restrict__ sumsq,
                          const float* __restrict__ gamma,
                          const float* __restrict__ beta,
                          float* __restrict__ bnsc, float* __restrict__ bnsh,
                          int n) {
  int c = threadIdx.x;
  if (c >= CH_HID) return;
  float inv_n = 1.0f / (float)n;
  float mean  = sums[c] * inv_n;
  float var   = sumsq[c] * inv_n - mean * mean;
  float sc    = gamma[c] * rsqrtf(var + BN_EPS);
  bnsc[c] = sc;
  bnsh[c] = beta[c] - mean * sc;
}

// ---------------------------------------------------------------------------
// 8) fused BN*ReLU*dot(W2): one wave per node, wave32 shuffle reduction
//    z[i] = sum_c relu(acc1[i,c]*sc[c]+sh[c]) * W2[c]
// ---------------------------------------------------------------------------
__global__ void k_dot(const float* __restrict__ acc,
                      const float* __restrict__ bnsc,
                      const float* __restrict__ bnsh,
                      const float* __restrict__ W2,
                      float* __restrict__ z, int n) {
  int wave = (blockIdx.x * blockDim.x + threadIdx.x) >> 5;
  int lane = threadIdx.x & 31;
  if (wave >= n) return;
  const float* row = acc + (size_t)wave * CH_HID;
  int c1 = lane, c2 = lane + 32;
  float v0 = fmaxf(fmaf(row[c1], bnsc[c1], bnsh[c1]), 0.f) * W2[c1];
  float v1 = fmaxf(fmaf(row[c2], bnsc[c2], bnsh[c2]), 0.f) * W2[c2];
  float s = v0 + v1;
#pragma unroll
  for (int off = 16; off > 0; off >>= 1) s += __shfl_xor(s, off, 32);
  if (lane == 0) z[wave] = s;
}

// ---------------------------------------------------------------------------
// 9) layer-2 aggregation (scalar): init with self loop + bias, then edges
// ---------------------------------------------------------------------------
__global__ void k_agg2_init(const float* __restrict__ z,
                            const float* __restrict__ dis,
                            const float* __restrict__ b2,
                            float* __restrict__ out2, int n) {
  int i = blockIdx.x * blockDim.x + threadIdx.x;
  if (i >= n) return;
  float d = dis[i];
  out2[i] = z[i] * d * d + b2[0];
}

__global__ void k_agg2_edges(const float* __restrict__ z,
                             const int* __restrict__ src,
                             const int* __restrict__ dst,
                             const float* __restrict__ dis,
                             float* __restrict__ out2, int nE) {
  int e = blockIdx.x * blockDim.x + threadIdx.x;
  if (e >= nE) return;
  int s = src[e], d = dst[e];
  atomicAdd(&out2[d], z[s] * dis[s] * dis[d]);
}

// ---------------------------------------------------------------------------
// 10) sigmoid -> d_out
// ---------------------------------------------------------------------------
__global__ void k_sigmoid(const float* __restrict__ out2,
                          float* __restrict__ out, int n) {
  int i = blockIdx.x * blockDim.x + threadIdx.x;
  if (i >= n) return;
  out[i] = 1.0f / (1.0f + expf(-out2[i]));
}

// ===========================================================================
extern "C" void kernel_launch(void* const* d_in, const int* in_sizes, int n_in,
                              void* d_out, int out_size, void* d_ws, size_t ws_size,
                              hipStream_t stream) {
  const float* x     = (const float*)d_in[0];
  const float* W1    = (const float*)d_in[1];
  const float* b1    = (const float*)d_in[2];
  const float* gamma = (const float*)d_in[3];
  const float* beta  = (const float*)d_in[4];
  const float* W2    = (const float*)d_in[5];
  const float* b2    = (const float*)d_in[6];
  const int*   eidx  = (const int*)d_in[7];   // [2, E] int32

  const int nNodes = in_sizes[0] / CH_IN;     // 100000
  const int nE     = in_sizes[7] / 2;         // 3200000
  const int* src = eidx;
  const int* dst = eidx + nE;

  // carve workspace (floats, 256B-aligned chunks)
  float* ws = (float*)d_ws;
  size_t off = 0;
  auto carve = [&](size_t nElem) {
    float* p = ws + off;
    off += (nElem + 63) & ~(size_t)63;
    return p;
  };
  float* deg   = carve(nNodes);
  float* dis   = carve(nNodes);
  float* hlin  = carve((size_t)nNodes * CH_HID);
  float* acc1  = carve((size_t)nNodes * CH_HID);
  float* sums  = carve(CH_HID);
  float* sumsq = carve(CH_HID);
  float* bnsc  = carve(CH_HID);
  float* bnsh  = carve(CH_HID);
  float* z     = carve(nNodes);
  float* out2  = carve(nNodes);
  (void)ws_size; (void)n_in; (void)out_size;

  const int B = 256;
  auto gridFor = [](long long total, int blk) { return (int)((total + blk - 1) / blk); };

  // 1-3: normalization coefficients
  k_init  <<<gridFor(nNodes, B), B, 0, stream>>>(deg, sums, sumsq, nNodes);
  k_degree<<<gridFor(nE, B),     B, 0, stream>>>(dst, deg, nE);
  k_dis   <<<gridFor(nNodes, B), B, 0, stream>>>(deg, dis, nNodes);

  // 4: WMMA GEMM  x @ W1  (+ fused self-loop/bias epilogue)
  k_gemm1_wmma<<<gridFor(nNodes, 32), B, 0, stream>>>(x, W1, dis, b1, hlin,
                                                      acc1, nNodes);

  // 5: layer-1 edge aggregation (L2-resident float atomics)
  k_agg1<<<gridFor((long long)nE * 16, B), B, 0, stream>>>(
      hlin, src, dst, dis, acc1, nE * 16);

  // 6-7: BatchNorm training statistics
  k_bnstats<<<1024, B, 0, stream>>>(acc1, sums, sumsq, nNodes);
  k_bnfinal<<<1, 64, 0, stream>>>(sums, sumsq, gamma, beta, bnsc, bnsh, nNodes);

  // 8: fused BN + ReLU + dot(W2) -> z
  k_dot<<<gridFor((long long)nNodes * 32, B), B, 0, stream>>>(acc1, bnsc, bnsh, W2, z, nNodes);

  // 9-10: layer-2 aggregation + sigmoid
  k_agg2_init <<<gridFor(nNodes, B), B, 0, stream>>>(z, dis, b2, out2, nNodes);
  k_agg2_edges<<<gridFor(nE, B),     B, 0, stream>>>(z, src, dst, dis, out2, nE);
  k_sigmoid   <<<gridFor(nNodes, B), B, 0, stream>>>(out2, (float*)d_out, nNodes);
}